// RGATLayer_51264729645650
// MI455X (gfx1250) — compile-verified
//
#include <hip/hip_runtime.h>
#include <hip/hip_bf16.h>

// ---------------------------------------------------------------------------
// RGAT layer, fused for MI455X (gfx1250, wave32).
// Memory-bound: 3 streaming passes over edge_h (410MB each) with the 64x64
// GEMM recomputed per pass via V_WMMA_F32_16X16X4_F32 (exact fp32).
// Node-sized scratch (m/denom/agg, 76.8MB) lives in d_ws and is L2-resident.
// ---------------------------------------------------------------------------

typedef __attribute__((ext_vector_type(2))) float v2f;
typedef __attribute__((ext_vector_type(8))) float v8f;

#define LDW_PITCH 68   // 64 + 4 pad floats -> conflict-free B-fragment reads

// Order-preserving int encoding of float for atomicMax-based segment max.
__device__ __forceinline__ int ordbits(float x) {
    int i = __float_as_int(x);
    return i >= 0 ? i : (i ^ 0x7fffffff);
}
__device__ __forceinline__ float unordbits(int o) {
    int i = o >= 0 ? o : (o ^ 0x7fffffff);
    return __int_as_float(i);
}

// Load 64x64 weight matrix into padded LDS.
// TR=false: lw[j][k] = W[j*64+k]   (for  X @ W.T : B[k][j] = W[j][k])
// TR=true : lw[j][k] = W[k*64+j]   (for  X @ W   : B[k][j] = W[k][j])
template <bool TR>
__device__ __forceinline__ void load_w_lds(float* lw, const float* __restrict__ W) {
    for (int i = threadIdx.x; i < 64 * 64; i += 256) {
        int r = i >> 6, c = i & 63;
        if (TR) lw[c * LDW_PITCH + r] = W[i];
        else    lw[r * LDW_PITCH + c] = W[i];
    }
    __syncthreads();
}

// One wave computes a 16x64 fp32 output tile: acc[t] (t=0..3) are 16x16
// accumulators. K=64 -> 16 chained V_WMMA_F32_16X16X4_F32 per col-tile.
// aptr = &A[rowA*64 + 2*half] (rowA = row0 + (lane&15), clamped by caller).
__device__ __forceinline__ void wmma_16x64(const float* __restrict__ aptr,
                                           const float* __restrict__ lw,
                                           int lm, int half, v8f acc[4]) {
#pragma unroll
    for (int s = 0; s < 16; ++s) {
        v2f a;
        a.x = aptr[4 * s + 0];   // A[M=lm][k0 + 2*half]
        a.y = aptr[4 * s + 1];   // A[M=lm][k0 + 2*half + 1]
#pragma unroll
        for (int t = 0; t < 4; ++t) {
            const float* bp = lw + (16 * t + lm) * LDW_PITCH + 4 * s + 2 * half;
            v2f b;
            b.x = bp[0];         // B[k0 + 2*half    ][N = 16t+lm]
            b.y = bp[1];         // B[k0 + 2*half + 1][N = 16t+lm]
            acc[t] = __builtin_amdgcn_wmma_f32_16x16x4_f32(
                false, a, false, b, (short)0, acc[t], false, false);
        }
    }
}

// ---------------------------------------------------------------------------
__global__ void __launch_bounds__(256) k_init(int* __restrict__ m_ord,
                                              float* __restrict__ denom,
                                              float* __restrict__ agg, int n) {
    int i = blockIdx.x * 256 + threadIdx.x;
    if (i < n) {
        m_ord[i] = (int)0x80000000;  // -inf in ordered encoding
        denom[i] = 0.0f;
        agg[i]   = 0.0f;
    }
}

// Y[row] = X[row] @ W.T  (tile GEMM, plain store) -- used for h = fc(node_h)
__global__ void __launch_bounds__(256) k_gemm_store(const float* __restrict__ X,
                                                    const float* __restrict__ W,
                                                    float* __restrict__ Y, int nrows) {
    __shared__ float lw[64 * LDW_PITCH];
    load_w_lds<false>(lw, W);
    int wave = threadIdx.x >> 5, lane = threadIdx.x & 31;
    int lm = lane & 15, half = lane >> 4;
    int row0 = (blockIdx.x * 8 + wave) * 16;
    if (row0 >= nrows) return;
    long long rowA = row0 + lm; if (rowA > nrows - 1) rowA = nrows - 1;
    v8f acc[4] = {};
    wmma_16x64(X + rowA * 64 + 2 * half, lw, lm, half, acc);
#pragma unroll
    for (int r = 0; r < 8; ++r) {
        int row = row0 + 8 * half + r;
        if (row < nrows) {
#pragma unroll
            for (int t = 0; t < 4; ++t)
                Y[(long long)row * 64 + 16 * t + lm] = acc[t][r];
        }
    }
}

// Pass 1: eh = edge_h @ W_fcr.T ; segment max into m_ord[dst] (ordered-int).
__global__ void __launch_bounds__(256) k_edge_max(const float* __restrict__ EH,
                                                  const float* __restrict__ W,
                                                  const int* __restrict__ dst,
                                                  int* __restrict__ m_ord, int E) {
    __shared__ float lw[64 * LDW_PITCH];
    load_w_lds<false>(lw, W);
    int wave = threadIdx.x >> 5, lane = threadIdx.x & 31;
    int lm = lane & 15, half = lane >> 4;
    int row0 = (blockIdx.x * 8 + wave) * 16;
    if (row0 >= E) return;
    long long rowA = row0 + lm; if (rowA > E - 1) rowA = E - 1;
    v8f acc[4] = {};
    wmma_16x64(EH + rowA * 64 + 2 * half, lw, lm, half, acc);
#pragma unroll
    for (int r = 0; r < 8; ++r) {
        int row = row0 + 8 * half + r;
        if (row < E) {
            int base = dst[row] * 64 + lm;
#pragma unroll
            for (int t = 0; t < 4; ++t)
                atomicMax(&m_ord[base + 16 * t], ordbits(acc[t][r]));
        }
    }
}

// Pass 2: recompute eh ; denom[dst] += exp(eh - m[dst]).
__global__ void __launch_bounds__(256) k_edge_denom(const float* __restrict__ EH,
                                                    const float* __restrict__ W,
                                                    const int* __restrict__ dst,
                                                    const int* __restrict__ m_ord,
                                                    float* __restrict__ denom, int E) {
    __shared__ float lw[64 * LDW_PITCH];
    load_w_lds<false>(lw, W);
    int wave = threadIdx.x >> 5, lane = threadIdx.x & 31;
    int lm = lane & 15, half = lane >> 4;
    int row0 = (blockIdx.x * 8 + wave) * 16;
    if (row0 >= E) return;
    long long rowA = row0 + lm; if (rowA > E - 1) rowA = E - 1;
    v8f acc[4] = {};
    wmma_16x64(EH + rowA * 64 + 2 * half, lw, lm, half, acc);
#pragma unroll
    for (int r = 0; r < 8; ++r) {
        int row = row0 + 8 * half + r;
        if (row < E) {
            int base = dst[row] * 64 + lm;
#pragma unroll
            for (int t = 0; t < 4; ++t) {
                float m = unordbits(m_ord[base + 16 * t]);
                atomicAdd(&denom[base + 16 * t], __expf(acc[t][r] - m));
            }
        }
    }
}

// Pass 3: recompute eh ; alpha = exp(eh-m)/denom ; agg[dst] += alpha*(h[src]+eh).
__global__ void __launch_bounds__(256) k_edge_scatter(const float* __restrict__ EH,
                                                      const float* __restrict__ W,
                                                      const int* __restrict__ src,
                                                      const int* __restrict__ dst,
                                                      const int* __restrict__ m_ord,
                                                      const float* __restrict__ denom,
                                                      const float* __restrict__ H,
                                                      float* __restrict__ agg, int E) {
    __shared__ float lw[64 * LDW_PITCH];
    load_w_lds<false>(lw, W);
    int wave = threadIdx.x >> 5, lane = threadIdx.x & 31;
    int lm = lane & 15, half = lane >> 4;
    int row0 = (blockIdx.x * 8 + wave) * 16;
    if (row0 >= E) return;
    long long rowA = row0 + lm; if (rowA > E - 1) rowA = E - 1;
    v8f acc[4] = {};
    wmma_16x64(EH + rowA * 64 + 2 * half, lw, lm, half, acc);
#pragma unroll
    for (int r = 0; r < 8; ++r) {
        int row = row0 + 8 * half + r;
        if (row < E) {
            int base = dst[row] * 64 + lm;
            long long sb = (long long)src[row] * 64 + lm;
#pragma unroll
            for (int t = 0; t < 4; ++t) {
                float v  = acc[t][r];
                float m  = unordbits(m_ord[base + 16 * t]);
                float al = __expf(v - m) / denom[base + 16 * t];
                float z  = H[sb + 16 * t];               // h[src] gather (L2-resident)
                atomicAdd(&agg[base + 16 * t], al * (z + v));
            }
        }
    }
}

// Final: out = relu( denom>0 ? agg + h@loop_W : h ), in place over H (=d_out).
__global__ void __launch_bounds__(256) k_final(float* __restrict__ H,
                                               const float* __restrict__ loopW,
                                               const float* __restrict__ denom,
                                               const float* __restrict__ agg, int N) {
    __shared__ float lw[64 * LDW_PITCH];
    load_w_lds<true>(lw, loopW);   // B[k][j] = loop_W[k][j]  (no transpose in math)
    int wave = threadIdx.x >> 5, lane = threadIdx.x & 31;
    int lm = lane & 15, half = lane >> 4;
    int row0 = (blockIdx.x * 8 + wave) * 16;
    if (row0 >= N) return;
    long long rowA = row0 + lm; if (rowA > N - 1) rowA = N - 1;
    v8f acc[4] = {};
    wmma_16x64(H + rowA * 64 + 2 * half, lw, lm, half, acc);  // h @ loop_W tile
#pragma unroll
    for (int r = 0; r < 8; ++r) {
        int row = row0 + 8 * half + r;
        if (row < N) {
#pragma unroll
            for (int t = 0; t < 4; ++t) {
                long long idx = (long long)row * 64 + 16 * t + lm;
                float hv = H[idx];
                float de = denom[idx];
                float o  = (de > 0.0f) ? (agg[idx] + acc[t][r]) : hv;
                H[idx] = fmaxf(o, 0.0f);
            }
        }
    }
}

// ---------------------------------------------------------------------------
extern "C" void kernel_launch(void* const* d_in, const int* in_sizes, int n_in,
                              void* d_out, int out_size, void* d_ws, size_t ws_size,
                              hipStream_t stream) {
    const float* node_h = (const float*)d_in[0];
    const float* edge_h = (const float*)d_in[1];
    const int*   src    = (const int*)d_in[2];
    const int*   dst    = (const int*)d_in[3];
    const float* W_fc   = (const float*)d_in[4];
    const float* W_fcr  = (const float*)d_in[5];
    const float* loop_W = (const float*)d_in[6];

    const int N = in_sizes[0] / 64;
    const int E = in_sizes[2];
    float* out = (float*)d_out;          // holds h after k_gemm_store, result after k_final

    // ws layout: m_ord | denom | agg, each N*64 (L2-resident: 3 * 25.6MB)
    char* ws = (char*)d_ws;
    size_t nd = (size_t)N * 64;
    int*   m_ord = (int*)ws;
    float* denom = (float*)(ws + nd * 4);
    float* agg   = (float*)(ws + 2 * nd * 4);

    const int tot = (int)nd;
    dim3 gInit((tot + 255) / 256);
    dim3 gN((N + 127) / 128);            // 8 waves/block * 16 rows/wave
    dim3 gE((E + 127) / 128);

    k_init<<<gInit, 256, 0, stream>>>(m_ord, denom, agg, tot);
    k_gemm_store<<<gN, 256, 0, stream>>>(node_h, W_fc, out, N);          // h = fc(node_h)
    k_edge_max<<<gE, 256, 0, stream>>>(edge_h, W_fcr, dst, m_ord, E);    // segment max
    k_edge_denom<<<gE, 256, 0, stream>>>(edge_h, W_fcr, dst, m_ord, denom, E);
    k_edge_scatter<<<gE, 256, 0, stream>>>(edge_h, W_fcr, src, dst, m_ord, denom,
                                           out, agg, E);
    k_final<<<gN, 256, 0, stream>>>(out, loop_W, denom, agg, N);
}